// My_HuberLoss_37976100831940
// MI455X (gfx1250) — compile-verified
//
#include <hip/hip_runtime.h>
#include <hip/hip_bf16.h>

typedef __attribute__((ext_vector_type(2))) float v2f;
typedef __attribute__((ext_vector_type(4))) float v4f;
typedef __attribute__((ext_vector_type(8))) float v8f;

#define DS_SWAPX16 0x401F  // group-of-32 swizzle: xor_mask=0x10, and_mask=0x1F

// Per-element loss:
//   r = |t - p|
//   r >= 5           : 5r - 12.5
//   r < 5            : (r < 2 && same_bucket(p,t) ? 0.05 : 0.5) * r^2
__device__ __forceinline__ float huber_elem(float p, float t) {
    float r = fabsf(t - p);
    bool sb = (p >= 0.0f  && p <= 13.0f && t >= 0.0f  && t <= 13.0f) ||
              (p >= 14.0f && p <= 19.0f && t >= 14.0f && t <= 19.0f) ||
              (p >= 20.0f && p <= 28.0f && t >= 20.0f && t <= 28.0f) ||
              (p >= 29.0f && p <= 63.0f && t >= 29.0f && t <= 63.0f);
    float coef     = (r < 2.0f && sb) ? 0.05f : 0.5f;
    float loss_in  = coef * r * r;
    float loss_out = 5.0f * r - 12.5f;
    return (r < 5.0f) ? loss_in : loss_out;
}

// Full 32-lane (wave32) sum via one V_WMMA_F32_16X16X4_F32 + one ds_swizzle.
// A-layout (32-bit 16x4): lanes 0-15 VGPR0 = K=0, lanes 16-31 VGPR0 = K=2.
// With a = {s, 0} and B = ones(4x16):  D[m][n] = s_m + s_(m+16) for all n.
// D-layout: lane L<16 holds D[0..7][L] in its 8 VGPRs, lane L>=16 holds D[8..15][L-16].
// Sum of own 8 D regs -> half-wave sum; swap halves (xor 16) + add -> full wave sum.
// Requires EXEC == all 1s (call sites are divergence-free).
__device__ __forceinline__ float wave_reduce_wmma(float s) {
    v2f a; a[0] = s;    a[1] = 0.0f;
    v2f b; b[0] = 1.0f; b[1] = 1.0f;   // all-ones B: layout-independent
    v8f c = {};
    v8f d = __builtin_amdgcn_wmma_f32_16x16x4_f32(
        /*neg_a=*/false, a, /*neg_b=*/false, b,
        /*c_mod=*/(short)0, c, /*reuse_a=*/false, /*reuse_b=*/false);
    float v = ((d[0] + d[1]) + (d[2] + d[3])) + ((d[4] + d[5]) + (d[6] + d[7]));
    int other = __builtin_amdgcn_ds_swizzle(__float_as_int(v), DS_SWAPX16);
    return v + __int_as_float(other);
}

// Kernel 1: each block reduces a contiguous 4096-element chunk -> one partial.
// 256 threads x 4 float4 per array, coalesced (thread t hits t, t+256, t+512, t+768).
__global__ __launch_bounds__(256) void huber_partial_kernel(
    const float* __restrict__ y_pred, const float* __restrict__ y_true,
    float* __restrict__ partials) {
    __shared__ float sm[8];
    const int tid = threadIdx.x;
    const v4f* __restrict__ pp = (const v4f*)y_pred;
    const v4f* __restrict__ tt = (const v4f*)y_true;
    const int base = blockIdx.x * 1024 + tid;   // in float4 units; 1024 float4 per block

    float s = 0.0f;
#pragma unroll
    for (int i = 0; i < 4; ++i) {
        v4f p = __builtin_nontemporal_load(pp + base + i * 256);
        v4f t = __builtin_nontemporal_load(tt + base + i * 256);
        s += huber_elem(p[0], t[0]);
        s += huber_elem(p[1], t[1]);
        s += huber_elem(p[2], t[2]);
        s += huber_elem(p[3], t[3]);
    }

    float ws = wave_reduce_wmma(s);          // all 8 waves fully active
    if ((tid & 31) == 0) sm[tid >> 5] = ws;
    __syncthreads();
    if (tid == 0) {
        float bsum = ((sm[0] + sm[1]) + (sm[2] + sm[3])) +
                     ((sm[4] + sm[5]) + (sm[6] + sm[7]));
        partials[blockIdx.x] = bsum;
    }
}

// Kernel 2: single block of 1024 threads deterministically reduces the
// per-block partials and writes mean = sum / N.
__global__ __launch_bounds__(1024) void huber_final_kernel(
    const float* __restrict__ partials, float* __restrict__ out,
    int n_partials, float inv_n) {
    __shared__ float sm[32];
    const int tid = threadIdx.x;

    float s = 0.0f;
    for (int i = tid; i < n_partials; i += 1024) s += partials[i];

    float ws = wave_reduce_wmma(s);          // all 32 waves fully active
    if ((tid & 31) == 0) sm[tid >> 5] = ws;
    __syncthreads();

    // Every wave redundantly reduces the 32 wave-sums (keeps EXEC all-1s for WMMA).
    float v = sm[tid & 31];
    float tot = wave_reduce_wmma(v);
    if (tid == 0) out[0] = tot * inv_n;
}

extern "C" void kernel_launch(void* const* d_in, const int* in_sizes, int n_in,
                              void* d_out, int out_size, void* d_ws, size_t ws_size,
                              hipStream_t stream) {
    const float* y_pred = (const float*)d_in[0];
    const float* y_true = (const float*)d_in[1];
    float* out      = (float*)d_out;
    float* partials = (float*)d_ws;

    const int n = in_sizes[0];                 // 16,777,216
    const int elems_per_block = 4096;          // 256 threads x 16 elems
    const int num_blocks = n / elems_per_block;// 4096 (N is exactly 4096^2)
    const float inv_n = 1.0f / (float)n;

    huber_partial_kernel<<<num_blocks, 256, 0, stream>>>(y_pred, y_true, partials);
    huber_final_kernel<<<1, 1024, 0, stream>>>(partials, out, num_blocks, inv_n);
}